// localCnn_2705829396574
// MI455X (gfx1250) — compile-verified
//
#include <hip/hip_runtime.h>

// LocallyConnected2D(3x3, valid) + BatchNorm(inference, eps=1e-3) + ReLU
// x:      [32, 32, 32, 64]  f32
// kernel: [30, 30, 576, 64] f32   (per-location unshared weights)
// bias:   [30, 30, 64], gamma/beta/mean/var: [64]
// out:    [32, 30, 30, 64]  f32
//
// One wave32 per output location (h,w): GEMM M=32(batch) x K=576 x N=64(filters)
// via V_WMMA_F32_16X16X4_F32. C-tile = 2 (M) x 4 (N) v8f accumulators.

#define BB   32
#define HH   32
#define WWi  32
#define CC   64
#define FF   64
#define OHH  30
#define OWW  30
#define KK   576
#define BN_EPS 1e-3f

typedef __attribute__((ext_vector_type(2))) float v2f;
typedef __attribute__((ext_vector_type(8))) float v8f;

__global__ __launch_bounds__(128) void lc2d_bn_relu_wmma(
    const float* __restrict__ x,
    const float* __restrict__ kern,
    const float* __restrict__ bias,
    const float* __restrict__ gamma,
    const float* __restrict__ beta,
    const float* __restrict__ mmean,
    const float* __restrict__ mvar,
    float* __restrict__ out)
{
    const int wavesPerBlock = blockDim.x >> 5;
    const int wave = blockIdx.x * wavesPerBlock + ((int)threadIdx.x >> 5);
    if (wave >= OHH * OWW) return;              // 225 blocks x 4 waves == 900: never diverges

    const int h    = wave / OWW;
    const int w    = wave % OWW;
    const int lane = (int)threadIdx.x & 31;
    const int half = lane >> 4;                 // 0: lanes 0-15, 1: lanes 16-31
    const int lm   = lane & 15;

    // D = A x B + C accumulators: acc[mtile][ntile], each 16x16 f32 in 8 VGPRs
    v8f acc[2][4];
    #pragma unroll
    for (int mt = 0; mt < 2; ++mt)
        #pragma unroll
        for (int t = 0; t < 4; ++t)
            acc[mt][t] = (v8f){0.f, 0.f, 0.f, 0.f, 0.f, 0.f, 0.f, 0.f};

    const float* wloc = kern + (size_t)(h * OWW + w) * (KK * FF);

    // K = 576 = 9 kernel positions x 64 channels; step K by 4 per WMMA.
    for (int kp = 0; kp < 9; ++kp) {
        const int ki = kp / 3;
        const int kj = kp % 3;
        // A-matrix rows: M = batch. lane L<16 holds M=L (K pair {k,k+1});
        // lane L>=16 holds M=L-16 (K pair {k+2,k+3}).
        const float* xrow0 = x + (((size_t)lm        * HH + (h + ki)) * WWi + (w + kj)) * CC;
        const float* xrow1 = x + (((size_t)(lm + 16) * HH + (h + ki)) * WWi + (w + kj)) * CC;
        const float* wk    = wloc + (size_t)kp * 64 * FF;   // rows k = kp*64 ..

        #pragma unroll 4
        for (int c0 = 0; c0 < 64; c0 += 4) {
            // A fragments: channels {c0+2*half, c0+2*half+1} -> contiguous 8B
            v2f a0 = *(const v2f*)(xrow0 + c0 + 2 * half);
            v2f a1 = *(const v2f*)(xrow1 + c0 + 2 * half);
            // B fragment base: row k = c0 + 2*half (this lane's K pair start)
            const float* wkk = wk + (size_t)(c0 + 2 * half) * FF;

            #pragma unroll
            for (int t = 0; t < 4; ++t) {
                // B 4x16: VGPR0 = B[k][n] (lanes0-15) / B[k+2][n] (lanes16-31),
                //         VGPR1 = B[k+1][n] / B[k+3][n];  n = t*16 + lm
                v2f b;
                b.x = wkk[      t * 16 + lm];   // lanes 0-15: 64B contiguous in f
                b.y = wkk[FF +  t * 16 + lm];
                acc[0][t] = __builtin_amdgcn_wmma_f32_16x16x4_f32(
                    false, a0, false, b, (short)0, acc[0][t], false, false);
                acc[1][t] = __builtin_amdgcn_wmma_f32_16x16x4_f32(
                    false, a1, false, b, (short)0, acc[1][t], false, false);
            }
        }
    }

    // Epilogue: y = dot*inv + (bias - mean)*inv + beta, then ReLU.
    // C/D layout: VGPR r, lane L -> M = 8*(L/16) + r, N = L%16.
    const float* bloc = bias + (size_t)(h * OWW + w) * FF;
    #pragma unroll
    for (int t = 0; t < 4; ++t) {
        const int f = t * 16 + lm;
        const float inv = gamma[f] * __frsqrt_rn(mvar[f] + BN_EPS);
        const float off = (bloc[f] - mmean[f]) * inv + beta[f];
        #pragma unroll
        for (int mt = 0; mt < 2; ++mt) {
            #pragma unroll
            for (int r = 0; r < 8; ++r) {
                const int b = mt * 16 + half * 8 + r;
                float v = acc[mt][t][r] * inv + off;
                v = v > 0.f ? v : 0.f;
                // lanes 0-15 of a given r share b -> 64B contiguous store in f
                out[(((size_t)b * OHH + h) * OWW + w) * FF + f] = v;
            }
        }
    }
}

extern "C" void kernel_launch(void* const* d_in, const int* in_sizes, int n_in,
                              void* d_out, int out_size, void* d_ws, size_t ws_size,
                              hipStream_t stream) {
    const float* x     = (const float*)d_in[0];
    const float* kern  = (const float*)d_in[1];
    const float* bias  = (const float*)d_in[2];
    const float* gamma = (const float*)d_in[3];
    const float* beta  = (const float*)d_in[4];
    const float* mmean = (const float*)d_in[5];
    const float* mvar  = (const float*)d_in[6];
    float* out = (float*)d_out;

    // 900 locations, 1 wave each, 4 waves (128 threads) per block -> 225 blocks
    dim3 grid(225), block(128);
    lc2d_bn_relu_wmma<<<grid, block, 0, stream>>>(x, kern, bias, gamma, beta,
                                                  mmean, mvar, out);
    (void)in_sizes; (void)n_in; (void)out_size; (void)d_ws; (void)ws_size;
}